// MultiHeadAttention_69990787056330
// MI455X (gfx1250) — compile-verified
//
#include <hip/hip_runtime.h>

typedef __attribute__((ext_vector_type(2))) float v2f;
typedef __attribute__((ext_vector_type(8))) float v8f;

// D = A(16x4) * B(4x16) + C(16x16), FP32 matrix pipe
#define WMMA_F32(a, b, c) \
  __builtin_amdgcn_wmma_f32_16x16x4_f32(false, (a), false, (b), (short)0, (c), false, false)

// ---------------------------------------------------------------------------
// GEMM: C[M,1024] = A[M,1024] * B[1024,1024]^T + bias  (K = N = 1024 fixed)
// Wave tile: 32(M) x 64(N). Block: 8 waves stacked in M => 256(M) x 64(N).
// Software-pipelined: chunk k+4 loads issued while chunk k feeds 8 WMMAs.
// ---------------------------------------------------------------------------
__global__ __launch_bounds__(256) void gemm_nt_kernel(
    const float* __restrict__ A, const float* __restrict__ B,
    const float* __restrict__ bias, float* __restrict__ C)
{
  const int Kd = 1024, Nd = 1024;
  const int lane = threadIdx.x & 31;
  const int w    = threadIdx.x >> 5;
  const int lr   = lane & 15;
  const int hf   = lane >> 4;

  const int m0 = blockIdx.y * 256 + w * 32;
  const int n0 = blockIdx.x * 64;

  const float* pa0 = A + (size_t)(m0 + lr)      * Kd + hf * 2;
  const float* pa1 = A + (size_t)(m0 + 16 + lr) * Kd + hf * 2;
  const float* pb0 = B + (size_t)(n0 + lr)      * Kd + hf * 2;
  const float* pb1 = pb0 + (size_t)16 * Kd;
  const float* pb2 = pb0 + (size_t)32 * Kd;
  const float* pb3 = pb0 + (size_t)48 * Kd;

  v8f acc[2][4];
#pragma unroll
  for (int r = 0; r < 2; ++r)
#pragma unroll
    for (int t = 0; t < 4; ++t) acc[r][t] = 0.0f;

  // Prologue: chunk 0 in flight.
  v2f a0 = *(const v2f*)(pa0);
  v2f a1 = *(const v2f*)(pa1);
  v2f b0 = *(const v2f*)(pb0);
  v2f b1 = *(const v2f*)(pb1);
  v2f b2 = *(const v2f*)(pb2);
  v2f b3 = *(const v2f*)(pb3);

  // Steady state: 255 iterations = 5 * 51.
#pragma unroll 5
  for (int k = 0; k < Kd - 4; k += 4) {
    v2f na0 = *(const v2f*)(pa0 + k + 4);
    v2f na1 = *(const v2f*)(pa1 + k + 4);
    v2f nb0 = *(const v2f*)(pb0 + k + 4);
    v2f nb1 = *(const v2f*)(pb1 + k + 4);
    v2f nb2 = *(const v2f*)(pb2 + k + 4);
    v2f nb3 = *(const v2f*)(pb3 + k + 4);
    acc[0][0] = WMMA_F32(a0, b0, acc[0][0]);
    acc[1][0] = WMMA_F32(a1, b0, acc[1][0]);
    acc[0][1] = WMMA_F32(a0, b1, acc[0][1]);
    acc[1][1] = WMMA_F32(a1, b1, acc[1][1]);
    acc[0][2] = WMMA_F32(a0, b2, acc[0][2]);
    acc[1][2] = WMMA_F32(a1, b2, acc[1][2]);
    acc[0][3] = WMMA_F32(a0, b3, acc[0][3]);
    acc[1][3] = WMMA_F32(a1, b3, acc[1][3]);
    a0 = na0; a1 = na1; b0 = nb0; b1 = nb1; b2 = nb2; b3 = nb3;
  }
  // Epilogue chunk.
  acc[0][0] = WMMA_F32(a0, b0, acc[0][0]);
  acc[1][0] = WMMA_F32(a1, b0, acc[1][0]);
  acc[0][1] = WMMA_F32(a0, b1, acc[0][1]);
  acc[1][1] = WMMA_F32(a1, b1, acc[1][1]);
  acc[0][2] = WMMA_F32(a0, b2, acc[0][2]);
  acc[1][2] = WMMA_F32(a1, b2, acc[1][2]);
  acc[0][3] = WMMA_F32(a0, b3, acc[0][3]);
  acc[1][3] = WMMA_F32(a1, b3, acc[1][3]);

#pragma unroll
  for (int t = 0; t < 4; ++t) {
    const float bv = bias[n0 + t * 16 + lr];
#pragma unroll
    for (int r = 0; r < 2; ++r) {
#pragma unroll
      for (int v = 0; v < 8; ++v) {
        const int row = m0 + r * 16 + hf * 8 + v;
        C[(size_t)row * Nd + n0 + t * 16 + lr] = acc[r][t][v] + bv;
      }
    }
  }
}

// ---------------------------------------------------------------------------
// Fused flash attention per (n, h). One wave owns 16 query rows, streams all
// 2048 keys in 16-key tiles with online softmax. S^T = K*Q^T trick keeps the
// softmax row stats scalar per lane. K and V tiles are batch-loaded into
// registers at the top of each tile so loads overlap WMMA + softmax VALU.
// ---------------------------------------------------------------------------
__global__ __launch_bounds__(256) void attn_kernel(
    const float* __restrict__ Q, const float* __restrict__ K,
    const float* __restrict__ V, const int* __restrict__ mask,
    float* __restrict__ O)
{
  const int T = 2048, D = 1024;
  __shared__ float pbuf[8 * 16 * 18];   // per-wave 16x16 P tile, [query][key], stride 18

  const int lane = threadIdx.x & 31;
  const int w    = threadIdx.x >> 5;
  const int lr   = lane & 15;
  const int hf   = lane >> 4;

  const int nh = blockIdx.x;
  const int n  = nh >> 4;
  const int h  = nh & 15;
  const int qb = blockIdx.y * 128 + w * 16;

  const size_t base = (size_t)n * T * D + (size_t)h * 64;
  const float* Qb = Q + base;
  const float* Kb = K + base;
  const float* Vb = V + base;
  float*       Ob = O + base;
  float* pl = &pbuf[w * 288];

  // Preload 16x64 Q block (Q^T B-chunks): float2 per chunk.
  v2f qa[16];
  const float* pq = Qb + (size_t)(qb + lr) * D + hf * 2;
#pragma unroll
  for (int c = 0; c < 16; ++c) qa[c] = *(const v2f*)(pq + 4 * c);

  // Per-lane query mask (query = qb + lr; duplicated across halves).
  const bool live = mask[n * T + qb + lr] != 0;

  v8f ot[4];
#pragma unroll
  for (int t = 0; t < 4; ++t) ot[t] = 0.0f;
  float m = -1e30f, l = 0.0f;

  const float inv_scale = 0.125f;          // 1/sqrt(64)
  const float NEGS = -1e20f * 0.125f;

  for (int kb = 0; kb < T; kb += 16) {
    // Batch-load K tile (A-chunks) and V tile (B-chunks) up front.
    const float* pk = Kb + (size_t)(kb + lr) * D + hf * 2;
    v2f kk[16];
#pragma unroll
    for (int c = 0; c < 16; ++c) kk[c] = *(const v2f*)(pk + 4 * c);

    const float* pv = Vb + (size_t)(kb + 2 * hf) * D + lr;
    v2f vv[4][4];
#pragma unroll
    for (int c = 0; c < 4; ++c)
#pragma unroll
      for (int t = 0; t < 4; ++t) {
        vv[c][t].x = pv[(4 * c)     * D + t * 16];
        vv[c][t].y = pv[(4 * c + 1) * D + t * 16];
      }

    // S^T tile (16 key x 16 query): lane holds (key = v + hf*8, query = lr).
    v8f s = 0.0f;
#pragma unroll
    for (int c = 0; c < 16; ++c) s = WMMA_F32(kk[c], qa[c], s);

    // Masked scale.
    float x[8];
#pragma unroll
    for (int v = 0; v < 8; ++v) x[v] = live ? s[v] * inv_scale : NEGS;

    // Row (query) max: 7 in-lane + 1 cross-half.
    float tm = x[0];
#pragma unroll
    for (int v = 1; v < 8; ++v) tm = fmaxf(tm, x[v]);
    tm = fmaxf(tm, __shfl_xor(tm, 16));

    const float mn = fmaxf(m, tm);
    const float al = __expf(m - mn);
    m = mn;

    float p[8];
    float ts = 0.0f;
#pragma unroll
    for (int v = 0; v < 8; ++v) { p[v] = __expf(x[v] - mn); ts += p[v]; }
    ts += __shfl_xor(ts, 16);
    l = l * al + ts;

    // Store P^T -> LDS as [query][key]: lane writes keys (hf*8+2j, +1).
#pragma unroll
    for (int j = 0; j < 4; ++j) {
      v2f pp; pp.x = p[2 * j]; pp.y = p[2 * j + 1];
      *(v2f*)&pl[lr * 18 + hf * 8 + 2 * j] = pp;
    }
    asm volatile("s_wait_dscnt 0" ::: "memory");  // cross-lane LDS RAW inside wave

    // Broadcast alpha to O-tile row mapping (row = v + hf*8).
    float av[8];
#pragma unroll
    for (int v = 0; v < 8; ++v) av[v] = __shfl(al, v + hf * 8);
#pragma unroll
    for (int t = 0; t < 4; ++t)
#pragma unroll
      for (int v = 0; v < 8; ++v) ot[t][v] *= av[v];

    // Read P back in A-layout (M=query, K=key), 4 chunks of K=4.
    v2f pa[4];
#pragma unroll
    for (int c = 0; c < 4; ++c)
      pa[c] = *(const v2f*)&pl[lr * 18 + 4 * c + hf * 2];

    // O += P * V.
#pragma unroll
    for (int c = 0; c < 4; ++c)
#pragma unroll
      for (int t = 0; t < 4; ++t)
        ot[t] = WMMA_F32(pa[c], vv[c][t], ot[t]);
  }

  // Normalize (broadcast 1/l to row mapping) and write ctx.
  const float linv = 1.0f / l;
  float lv[8];
#pragma unroll
  for (int v = 0; v < 8; ++v) lv[v] = __shfl(linv, v + hf * 8);
#pragma unroll
  for (int t = 0; t < 4; ++t)
#pragma unroll
    for (int v = 0; v < 8; ++v)
      Ob[(size_t)(qb + hf * 8 + v) * D + t * 16 + lr] = ot[t][v] * lv[v];
}

// ---------------------------------------------------------------------------
extern "C" void kernel_launch(void* const* d_in, const int* in_sizes, int n_in,
                              void* d_out, int out_size, void* d_ws, size_t ws_size,
                              hipStream_t stream) {
  const float* query = (const float*)d_in[0];
  const int*   mask  = (const int*)d_in[1];
  const float* Wq = (const float*)d_in[2]; const float* bq = (const float*)d_in[3];
  const float* Wk = (const float*)d_in[4]; const float* bk = (const float*)d_in[5];
  const float* Wv = (const float*)d_in[6]; const float* bv = (const float*)d_in[7];
  const float* Wo = (const float*)d_in[8]; const float* bo = (const float*)d_in[9];
  float* out = (float*)d_out;

  const int Mrows = 4 * 2048;       // N*T
  const int Dm = 1024;
  const size_t NTD = (size_t)Mrows * Dm;

  float* ws = (float*)d_ws;
  float* Qw = ws;
  float* Kw = ws + NTD;
  float* Vw = ws + 2 * NTD;
  float* Cw = ws + 3 * NTD;

  dim3 ggrid(Dm / 64, Mrows / 256);   // (16, 32)
  gemm_nt_kernel<<<ggrid, 256, 0, stream>>>(query, Wq, bq, Qw);
  gemm_nt_kernel<<<ggrid, 256, 0, stream>>>(query, Wk, bk, Kw);
  gemm_nt_kernel<<<ggrid, 256, 0, stream>>>(query, Wv, bv, Vw);

  attn_kernel<<<dim3(64, 16), 256, 0, stream>>>(Qw, Kw, Vw, mask, Cw);

  gemm_nt_kernel<<<ggrid, 256, 0, stream>>>(Cw, Wo, bo, out);
}